// DiffeqSolver_49203145343541
// MI455X (gfx1250) — compile-verified
//
#include <hip/hip_runtime.h>
#include <math.h>

typedef __attribute__((ext_vector_type(2))) float v2f;
typedef __attribute__((ext_vector_type(8))) float v8f;

// Problem dims (from reference)
#define S_ 4
#define B_ 1024
#define D_ 128     // state dim
#define H_ 256     // hidden dim
#define T_ 200

#define MROWS    32           // samples (rows) per workgroup
#define NTHREADS 512          // 16 wave32s
#define NWAVES   16

// LDS row strides in floats. stride % 64 (dwords) == 4 => conflict-free b64
// pattern for (row = lane%16, col = k0 + 2*(lane/16)) accesses, and even =>
// 8-byte alignment for v2f loads is preserved per row.
#define W1T_STRIDE 132        // 128 K + 4 pad
#define W2T_STRIDE 260        // 256 K + 4 pad
#define H_STRIDE   260
#define YT_STRIDE  132
#define KO_STRIDE  132

// float offsets into dynamic LDS
#define W1T_OFF 0
#define W2T_OFF (W1T_OFF + H_ * W1T_STRIDE)      // 33792
#define H_OFF   (W2T_OFF + D_ * W2T_STRIDE)      // 67072
#define KO_OFF  H_OFF                            // k-out aliases h (barrier-separated)
#define YT_OFF  (H_OFF + MROWS * H_STRIDE)       // 75392
#define B1_OFF  (YT_OFF + MROWS * YT_STRIDE)     // 79616
#define B2_OFF  (B1_OFF + H_)                    // 79872
#define LDS_FLOATS (B2_OFF + D_)                 // 80000 floats = 320000 B

// Native CDNA5 v_tanh_f32 if the device toolchain exposes it; otherwise a
// branch-free tanh on the hardware exp2/rcp (no exec divergence). The plain
// libm fallback exists only so the host compilation pass can parse this TU.
__device__ __forceinline__ float fast_tanh(float x) {
#if defined(__AMDGCN__) && __has_builtin(__builtin_amdgcn_tanhf)
    return __builtin_amdgcn_tanhf(x);
#elif defined(__AMDGCN__)
    // tanh(x) = 1 - 2/(exp(2x)+1), exp(2x) = exp2(x * 2*log2(e))
    float e = __builtin_amdgcn_exp2f(x * 2.8853900817779268147f);
    return 1.0f - 2.0f * __builtin_amdgcn_rcpf(e + 1.0f);
#else
    float e = exp2f(x * 2.8853900817779268147f);
    return 1.0f - 2.0f / (e + 1.0f);
#endif
}

__global__ __launch_bounds__(NTHREADS)
void node_rk4_kernel(const float* __restrict__ y0,
                     const float* __restrict__ ts,
                     const float* __restrict__ W1,
                     const float* __restrict__ b1g,
                     const float* __restrict__ W2,
                     const float* __restrict__ b2g,
                     float* __restrict__ out)
{
    extern __shared__ float lds[];
    float* w1t = lds + W1T_OFF;   // [H_][W1T_STRIDE]  w1t[n][k] = W1[k][n]
    float* w2t = lds + W2T_OFF;   // [D_][W2T_STRIDE]  w2t[n][k] = W2[k][n]
    float* hb  = lds + H_OFF;     // [MROWS][H_STRIDE] tanh hidden activations
    float* ko  = lds + KO_OFF;    // [MROWS][KO_STRIDE] f output (aliases hb)
    float* yt  = lds + YT_OFF;    // [MROWS][YT_STRIDE] current f input
    float* b1s = lds + B1_OFF;
    float* b2s = lds + B2_OFF;

    const unsigned tid   = threadIdx.x;
    const unsigned lane  = tid & 31u;
    const unsigned wave  = tid >> 5;
    const unsigned lhalf = lane >> 4;      // 0/1 : K sub-pair select
    const unsigned lmod  = lane & 15u;     // row/col within tile

    // ---------------- stage weights transposed into LDS (once) ----------------
    for (unsigned i = tid; i < (unsigned)(D_ * H_); i += NTHREADS) {
        unsigned k = i / H_, n = i % H_;
        w1t[n * W1T_STRIDE + k] = W1[i];
    }
    for (unsigned i = tid; i < (unsigned)(H_ * D_); i += NTHREADS) {
        unsigned k = i / D_, n = i % D_;
        w2t[n * W2T_STRIDE + k] = W2[i];
    }
    if (tid < H_) b1s[tid] = b1g[tid];
    if (tid < D_) b2s[tid] = b2g[tid];

    // ---------------- per-thread RK4 state: 8 contiguous elems of one row -----
    const unsigned r     = tid >> 4;            // 0..31 local row
    const unsigned c0    = (tid & 15u) * 8u;    // column base
    const unsigned gsrow = blockIdx.x * MROWS + r;

    float yreg[8];
    float accreg[8];
    {
        const float* yp = y0 + (size_t)gsrow * D_ + c0;
        #pragma unroll
        for (int j = 0; j < 8; ++j) yreg[j] = yp[j];
    }
    float* outrow = out + (size_t)gsrow * (size_t)(T_ * D_) + c0;
    #pragma unroll
    for (int j = 0; j < 8; ++j) outrow[j] = yreg[j];          // t = 0 slice
    #pragma unroll
    for (int j = 0; j < 8; ++j) yt[r * YT_STRIDE + c0 + j] = yreg[j];
    __syncthreads();

    // GEMM1 tiles: 2 per wave; GEMM2 tile: 1 per wave
    unsigned mt1[2], nt1[2];
    #pragma unroll
    for (int i = 0; i < 2; ++i) {
        unsigned tile = wave * 2 + (unsigned)i;
        mt1[i] = tile >> 4;
        nt1[i] = tile & 15u;
    }
    const unsigned mt2 = wave >> 3;
    const unsigned nt2 = wave & 7u;

    // precomputed LDS row bases (floats)
    unsigned aBase1[2], bBase1[2];
    #pragma unroll
    for (int i = 0; i < 2; ++i) {
        aBase1[i] = (mt1[i] * 16u + lmod) * YT_STRIDE + lhalf * 2u;
        bBase1[i] = (nt1[i] * 16u + lmod) * W1T_STRIDE + lhalf * 2u;
    }
    const unsigned aBase2 = (mt2 * 16u + lmod) * H_STRIDE   + lhalf * 2u;
    const unsigned bBase2 = (nt2 * 16u + lmod) * W2T_STRIDE + lhalf * 2u;

    const unsigned koRd = r * KO_STRIDE + c0;
    const unsigned ytWr = r * YT_STRIDE + c0;

    for (int step = 0; step < T_ - 1; ++step) {
        const float t0 = ts[step];
        const float t1 = ts[step + 1];
        const float hstep = t1 - t0;
        const float hh = 0.5f * hstep;

        #pragma unroll 1
        for (int sub = 0; sub < 4; ++sub) {
            // ---------------- GEMM1: h = yt @ W1 + b1, then tanh -------------
            // Two tiles x two K-chains each = 4 independent WMMA streams.
            v8f acc1a[2], acc1b[2];
            #pragma unroll
            for (int i = 0; i < 2; ++i) {
                float bb = b1s[nt1[i] * 16u + lmod];
                #pragma unroll
                for (int v = 0; v < 8; ++v) { acc1a[i][v] = bb; acc1b[i][v] = 0.0f; }
            }
            #pragma unroll 2
            for (int kk = 0; kk < D_ / 4; kk += 2) {
                unsigned k0a = (unsigned)kk * 4u;
                unsigned k0b = k0a + 4u;
                #pragma unroll
                for (int i = 0; i < 2; ++i) {
                    v2f a = *(const v2f*)&yt[aBase1[i] + k0a];
                    v2f b = *(const v2f*)&w1t[bBase1[i] + k0a];
                    acc1a[i] = __builtin_amdgcn_wmma_f32_16x16x4_f32(
                        false, a, false, b, (short)0, acc1a[i], false, false);
                }
                #pragma unroll
                for (int i = 0; i < 2; ++i) {
                    v2f a = *(const v2f*)&yt[aBase1[i] + k0b];
                    v2f b = *(const v2f*)&w1t[bBase1[i] + k0b];
                    acc1b[i] = __builtin_amdgcn_wmma_f32_16x16x4_f32(
                        false, a, false, b, (short)0, acc1b[i], false, false);
                }
            }
            #pragma unroll
            for (int i = 0; i < 2; ++i) {
                unsigned base = (mt1[i] * 16u + lhalf * 8u) * H_STRIDE + nt1[i] * 16u + lmod;
                #pragma unroll
                for (int v = 0; v < 8; ++v)
                    hb[base + (unsigned)v * H_STRIDE] = fast_tanh(acc1a[i][v] + acc1b[i][v]);
            }
            __syncthreads();

            // ---------------- GEMM2: k = h @ W2 + b2 -------------------------
            // Two independent K-chains.
            v8f acc2a, acc2b;
            {
                float bb = b2s[nt2 * 16u + lmod];
                #pragma unroll
                for (int v = 0; v < 8; ++v) { acc2a[v] = bb; acc2b[v] = 0.0f; }
            }
            #pragma unroll 2
            for (int kk = 0; kk < H_ / 4; kk += 2) {
                unsigned k0a = (unsigned)kk * 4u;
                unsigned k0b = k0a + 4u;
                {
                    v2f a = *(const v2f*)&hb[aBase2 + k0a];
                    v2f b = *(const v2f*)&w2t[bBase2 + k0a];
                    acc2a = __builtin_amdgcn_wmma_f32_16x16x4_f32(
                        false, a, false, b, (short)0, acc2a, false, false);
                }
                {
                    v2f a = *(const v2f*)&hb[aBase2 + k0b];
                    v2f b = *(const v2f*)&w2t[bBase2 + k0b];
                    acc2b = __builtin_amdgcn_wmma_f32_16x16x4_f32(
                        false, a, false, b, (short)0, acc2b, false, false);
                }
            }
            __syncthreads();   // all hb reads done before aliased ko writes
            {
                unsigned base = (mt2 * 16u + lhalf * 8u) * KO_STRIDE + nt2 * 16u + lmod;
                #pragma unroll
                for (int v = 0; v < 8; ++v)
                    ko[base + (unsigned)v * KO_STRIDE] = acc2a[v] + acc2b[v];
            }
            __syncthreads();

            // ---------------- elementwise RK4 combine ------------------------
            float kv[8];
            #pragma unroll
            for (int j = 0; j < 8; ++j) kv[j] = ko[koRd + j];

            if (sub == 0) {
                #pragma unroll
                for (int j = 0; j < 8; ++j) {
                    accreg[j] = kv[j];
                    yt[ytWr + j] = yreg[j] + hh * kv[j];
                }
            } else if (sub == 1) {
                #pragma unroll
                for (int j = 0; j < 8; ++j) {
                    accreg[j] += 2.0f * kv[j];
                    yt[ytWr + j] = yreg[j] + hh * kv[j];
                }
            } else if (sub == 2) {
                #pragma unroll
                for (int j = 0; j < 8; ++j) {
                    accreg[j] += 2.0f * kv[j];
                    yt[ytWr + j] = yreg[j] + hstep * kv[j];
                }
            } else {
                const float s6 = hstep * (1.0f / 6.0f);
                float* op = outrow + (size_t)(step + 1) * D_;
                #pragma unroll
                for (int j = 0; j < 8; ++j) {
                    accreg[j] += kv[j];
                    yreg[j] += s6 * accreg[j];
                    op[j] = yreg[j];
                    yt[ytWr + j] = yreg[j];
                }
            }
            __syncthreads();
        }
    }
}

extern "C" void kernel_launch(void* const* d_in, const int* in_sizes, int n_in,
                              void* d_out, int out_size, void* d_ws, size_t ws_size,
                              hipStream_t stream) {
    const float* y0 = (const float*)d_in[0];   // first_point (S,B,D)
    const float* ts = (const float*)d_in[1];   // time_steps (T)
    const float* W1 = (const float*)d_in[2];   // (D,H)
    const float* b1 = (const float*)d_in[3];   // (H)
    const float* W2 = (const float*)d_in[4];   // (H,D)
    const float* b2 = (const float*)d_in[5];   // (D)
    float* out = (float*)d_out;                // (S,B,T,D)

    dim3 grid((S_ * B_) / MROWS);   // 128 workgroups, each an independent ODE batch
    dim3 block(NTHREADS);
    size_t ldsBytes = (size_t)LDS_FLOATS * sizeof(float);   // 320000 B
    node_rk4_kernel<<<grid, block, ldsBytes, stream>>>(y0, ts, W1, b1, W2, b2, out);
}